// TransformerEncoder_33758442947013
// MI455X (gfx1250) — compile-verified
//
#include <hip/hip_runtime.h>
#include <hip/hip_bf16.h>
#include <math.h>

typedef float v2f __attribute__((ext_vector_type(2)));
typedef float v8f __attribute__((ext_vector_type(8)));

#define B_SZ 16
#define S_LEN 500
#define DMODEL 256
#define NHEAD 8
#define HDIM 32
#define FFDIM 1024
#define ROWS (B_SZ * S_LEN) /* 8000, multiple of 16 */

__device__ __forceinline__ float gelu_exact(float x) {
    return 0.5f * x * (1.0f + erff(x * 0.70710678118654752f));
}

// ---------------------------------------------------------------------------
// Generic fp32 WMMA GEMM:  C[ROWS x N] = act(A[ROWS x K] @ W[K x N] + bias)
//                                        (+ resid)
// block = 128 threads (4 waves); wave computes a 16(M) x 32(N) tile using
// V_WMMA_F32_16X16X4_F32 (two accumulators, A fragment reused).
// grid = (ROWS/16, ceil(N/128)). All N used here are multiples of 32 and
// never leave a partially-active wave, so EXEC is all-1s at every WMMA.
// ---------------------------------------------------------------------------
__global__ __launch_bounds__(128) void gemm_wmma_kernel(
    const float* __restrict__ A, const float* __restrict__ W,
    const float* __restrict__ bias, const float* resid, float* C,
    int K, int N, int act)
{
    const int wave  = threadIdx.x >> 5;
    const int lane  = threadIdx.x & 31;
    const int halfq = lane >> 4;   // lane half selects K pair (f32 WMMA layout)
    const int l16   = lane & 15;
    const int row0  = blockIdx.x << 4;
    const int col0  = blockIdx.y * 128 + wave * 32;
    if (col0 >= N) return;

    v8f acc0 = {0.f, 0.f, 0.f, 0.f, 0.f, 0.f, 0.f, 0.f};
    v8f acc1 = acc0;

    // A fragment (16x4): lane l16 = row M, halves hold K={0,1} / K={2,3}
    const float* Ap = A + (size_t)(row0 + l16) * K + 2 * halfq;
    // B fragment (4x16): lane l16 = col N, halves hold K={0,1} / K={2,3}
    const float* Wp = W + (size_t)(2 * halfq) * N + col0 + l16;

    #pragma unroll 2
    for (int k = 0; k < K; k += 4) {
        v2f a = *(const v2f*)(Ap + k);          // aligned: even float offset
        const float* wrow = Wp + (size_t)k * N;
        v2f b0 = { wrow[0],  wrow[N] };
        v2f b1 = { wrow[16], wrow[N + 16] };
        acc0 = __builtin_amdgcn_wmma_f32_16x16x4_f32(false, a, false, b0,
                                                     (short)0, acc0, false, false);
        acc1 = __builtin_amdgcn_wmma_f32_16x16x4_f32(false, a, false, b1,
                                                     (short)0, acc1, false, false);
    }

    // C/D layout: VGPR r -> M = r + 8*half, N = l16
    const int n0 = col0 + l16;
    const int n1 = n0 + 16;
    #pragma unroll
    for (int r = 0; r < 8; ++r) {
        int m = row0 + r + 8 * halfq;
        size_t base = (size_t)m * N;
        float v0 = acc0[r] + bias[n0];
        float v1 = acc1[r] + bias[n1];
        if (act) { v0 = gelu_exact(v0); v1 = gelu_exact(v1); }
        if (resid) { v0 += resid[base + n0]; v1 += resid[base + n1]; }
        C[base + n0] = v0;
        C[base + n1] = v1;
    }
}

// ---------------------------------------------------------------------------
// LayerNorm: one row per block (dim = 256 or 512). Safe for in == out.
// ---------------------------------------------------------------------------
__global__ __launch_bounds__(256) void ln_kernel(
    const float* in, const float* __restrict__ g, const float* __restrict__ bt,
    float* out, int dim, float eps)
{
    __shared__ float red[18];
    const int row = blockIdx.x;
    const float* x = in + (size_t)row * dim;
    float s = 0.f, s2 = 0.f;
    for (int i = threadIdx.x; i < dim; i += 256) { float v = x[i]; s += v; s2 += v * v; }
    #pragma unroll
    for (int o = 16; o > 0; o >>= 1) { s += __shfl_xor(s, o, 32); s2 += __shfl_xor(s2, o, 32); }
    const int wid = threadIdx.x >> 5;
    if ((threadIdx.x & 31) == 0) { red[wid] = s; red[8 + wid] = s2; }
    __syncthreads();
    if (threadIdx.x == 0) {
        float a = 0.f, b = 0.f;
        for (int i = 0; i < 8; ++i) { a += red[i]; b += red[8 + i]; }
        float mu  = a / dim;
        float var = b / dim - mu * mu;
        red[16] = mu;
        red[17] = rsqrtf(var + eps);
    }
    __syncthreads();
    const float mu = red[16], rs = red[17];
    float* o = out + (size_t)row * dim;
    for (int i = threadIdx.x; i < dim; i += 256)
        o[i] = (x[i] - mu) * rs * g[i] + bt[i];
}

// ---------------------------------------------------------------------------
// Patchify: P[(b*500+p)*512 + t*128 + f] = x[((4p+t)*16 + b)*128 + f]
// ---------------------------------------------------------------------------
__global__ __launch_bounds__(256) void patchify_kernel(
    const float* __restrict__ x, float* __restrict__ P)
{
    int idx = blockIdx.x * 256 + threadIdx.x;
    if (idx >= ROWS * 512) return;
    int f  = idx & 127;
    int t  = (idx >> 7) & 3;
    int rp = idx >> 9;              // b*500 + p
    int p  = rp % S_LEN;
    int b  = rp / S_LEN;
    P[idx] = x[((size_t)((4 * p + t) * B_SZ + b) << 7) + f];
}

// ---------------------------------------------------------------------------
// Causal multi-head attention with clipped relative bias.
// One block per (batch, head, 16-query tile). qkv rows are [q(256)|k(256)|v(256)].
// Scores staged in LDS (16x512), v staged in 128-row chunks.
// ---------------------------------------------------------------------------
__global__ __launch_bounds__(256) void attn_kernel(
    const float* __restrict__ qkv, const float* __restrict__ rel_bias,
    float* __restrict__ o)
{
    __shared__ float qs[16 * HDIM];
    __shared__ float sc[16 * 512];
    __shared__ float vs[128 * HDIM];

    const int tid = threadIdx.x;
    const int qt  = blockIdx.x & 31;         // 32 query tiles (last partial)
    const int hh  = (blockIdx.x >> 5) & 7;
    const int bb  = blockIdx.x >> 8;
    const int s0  = qt * 16;
    const int send = min(s0 + 16, S_LEN);    // max key index + 1 needed (causal)
    const float scale = 0.17677669529663687f; // 1/sqrt(32)

    // load + pre-scale q tile
    for (int i = tid; i < 16 * HDIM; i += 256) {
        int qi = i >> 5, d2 = i & 31;
        int s = s0 + qi;
        qs[i] = (s < S_LEN) ? qkv[((size_t)(bb * S_LEN + s)) * 768 + hh * HDIM + d2] * scale
                            : 0.f;
    }
    __syncthreads();

    // scores: sc[qi][kj] = q.k + bias, causal mask
    for (int kj = tid; kj < send; kj += 256) {
        float kv[HDIM];
        const float* kp = qkv + ((size_t)(bb * S_LEN + kj)) * 768 + DMODEL + hh * HDIM;
        #pragma unroll
        for (int d2 = 0; d2 < HDIM; ++d2) kv[d2] = kp[d2];
        #pragma unroll 4
        for (int qi = 0; qi < 16; ++qi) {
            int s = s0 + qi;
            float dot = 0.f;
            #pragma unroll
            for (int d2 = 0; d2 < HDIM; ++d2) dot += qs[(qi << 5) + d2] * kv[d2];
            int rel = kj - s;
            rel = rel < -64 ? -64 : (rel > 64 ? 64 : rel);
            float v = dot + rel_bias[hh * 129 + rel + 64];
            if (kj > s) v = -1.0e30f;        // causal
            sc[(qi << 9) + kj] = v;
        }
    }
    __syncthreads();

    // softmax per row (wave w handles rows w, w+8); zero masked tail for AV
    const int wid = tid >> 5, lane = tid & 31;
    for (int qi = wid; qi < 16; qi += 8) {
        int s = s0 + qi;
        if (s >= S_LEN) continue;
        int n = s + 1;
        float m = -1.0e30f;
        for (int j = lane; j < n; j += 32) m = fmaxf(m, sc[(qi << 9) + j]);
        #pragma unroll
        for (int off = 16; off > 0; off >>= 1) m = fmaxf(m, __shfl_xor(m, off, 32));
        float sum = 0.f;
        for (int j = lane; j < n; j += 32) {
            float e = expf(sc[(qi << 9) + j] - m);
            sc[(qi << 9) + j] = e;
            sum += e;
        }
        #pragma unroll
        for (int off = 16; off > 0; off >>= 1) sum += __shfl_xor(sum, off, 32);
        float inv = 1.f / sum;
        for (int j = lane; j < n; j += 32) sc[(qi << 9) + j] *= inv;
        for (int j = n + lane; j < send; j += 32) sc[(qi << 9) + j] = 0.f;
    }
    __syncthreads();

    // o = attn @ v, v staged in LDS chunks; thread owns (qiA,d) and (qiB,d)
    const int d2  = tid & 31;
    const int qiA = tid >> 5;
    const int qiB = qiA + 8;
    float acc0 = 0.f, acc1 = 0.f;
    for (int jc = 0; jc < send; jc += 128) {
        int jn = min(128, send - jc);
        for (int i = tid; i < jn * HDIM; i += 256) {
            int j = i >> 5, dd = i & 31;
            vs[i] = qkv[((size_t)(bb * S_LEN + jc + j)) * 768 + 2 * DMODEL + hh * HDIM + dd];
        }
        __syncthreads();
        for (int j = 0; j < jn; ++j) {
            float vv = vs[(j << 5) + d2];
            acc0 += sc[(qiA << 9) + jc + j] * vv;
            acc1 += sc[(qiB << 9) + jc + j] * vv;
        }
        __syncthreads();
    }
    int sA = s0 + qiA, sB = s0 + qiB;
    if (sA < S_LEN) o[((size_t)(bb * S_LEN + sA)) * DMODEL + hh * HDIM + d2] = acc0;
    if (sB < S_LEN) o[((size_t)(bb * S_LEN + sB)) * DMODEL + hh * HDIM + d2] = acc1;
}

// ---------------------------------------------------------------------------
// Final transpose: out[(s*16 + b)*128 + f] = src[(b*500 + s)*128 + f]
// ---------------------------------------------------------------------------
__global__ __launch_bounds__(256) void output_transpose_kernel(
    const float* __restrict__ src, float* __restrict__ dst)
{
    int idx = blockIdx.x * 256 + threadIdx.x;
    if (idx >= ROWS * 128) return;
    int f = idx & 127;
    int r = idx >> 7;
    int s = r % S_LEN;
    int b = r / S_LEN;
    dst[((size_t)(s * B_SZ + b) << 7) + f] = src[idx];
}

extern "C" void kernel_launch(void* const* d_in, const int* in_sizes, int n_in,
                              void* d_out, int out_size, void* d_ws, size_t ws_size,
                              hipStream_t stream)
{
    (void)in_sizes; (void)n_in; (void)out_size; (void)ws_size;
    const float* x        = (const float*)d_in[0];
    const float* ln1_g    = (const float*)d_in[1];
    const float* ln1_b    = (const float*)d_in[2];
    const float* proj_w   = (const float*)d_in[3];
    const float* proj_b   = (const float*)d_in[4];
    const float* ln2_g    = (const float*)d_in[5];
    const float* ln2_b    = (const float*)d_in[6];
    const float* rel_bias = (const float*)d_in[7];
    const float* lag      = (const float*)d_in[8];
    const float* lab      = (const float*)d_in[9];
    const float* qkv_w    = (const float*)d_in[10];
    const float* qkv_b    = (const float*)d_in[11];
    const float* out_w    = (const float*)d_in[12];
    const float* out_b    = (const float*)d_in[13];
    const float* lfg      = (const float*)d_in[14];
    const float* lfb      = (const float*)d_in[15];
    const float* ff1_w    = (const float*)d_in[16];
    const float* ff1_b    = (const float*)d_in[17];
    const float* ff2_w    = (const float*)d_in[18];
    const float* ff2_b    = (const float*)d_in[19];
    const float* fln_g    = (const float*)d_in[20];
    const float* fln_b    = (const float*)d_in[21];
    const float* op_w     = (const float*)d_in[22];
    const float* op_b     = (const float*)d_in[23];
    float* out = (float*)d_out;

    // Workspace layout (floats), aliased to minimize footprint (~74 MB):
    //   big : ROWS*1024  -- patch buffer (first ROWS*512) / ff1 out / attn out
    //   h   : ROWS*256   -- residual stream
    //   tmp : ROWS*256   -- LN scratch
    //   qkv : ROWS*768   -- qkv; reused as final-GEMM out (ROWS*128)
    float* ws  = (float*)d_ws;
    float* big = ws;
    float* h   = big + (size_t)ROWS * FFDIM;
    float* tmp = h   + (size_t)ROWS * DMODEL;
    float* qkv = tmp + (size_t)ROWS * DMODEL;

    // front end: patchify -> LN1 -> proj GEMM -> LN2
    patchify_kernel<<<(ROWS * 512 + 255) / 256, 256, 0, stream>>>(x, big);
    ln_kernel<<<ROWS, 256, 0, stream>>>(big, ln1_g, ln1_b, big, 512, 1e-6f);
    gemm_wmma_kernel<<<dim3(ROWS / 16, 2), 128, 0, stream>>>(
        big, proj_w, proj_b, nullptr, h, 512, DMODEL, 0);
    ln_kernel<<<ROWS, 256, 0, stream>>>(h, ln2_g, ln2_b, h, DMODEL, 1e-6f);

    for (int l = 0; l < 4; ++l) {
        ln_kernel<<<ROWS, 256, 0, stream>>>(h, lag + l * DMODEL, lab + l * DMODEL,
                                            tmp, DMODEL, 1e-5f);
        gemm_wmma_kernel<<<dim3(ROWS / 16, 6), 128, 0, stream>>>(
            tmp, qkv_w + (size_t)l * DMODEL * 3 * DMODEL, qkv_b + l * 3 * DMODEL,
            nullptr, qkv, DMODEL, 3 * DMODEL, 0);
        attn_kernel<<<B_SZ * NHEAD * 32, 256, 0, stream>>>(qkv, rel_bias, big);
        gemm_wmma_kernel<<<dim3(ROWS / 16, 2), 128, 0, stream>>>(
            big, out_w + (size_t)l * DMODEL * DMODEL, out_b + l * DMODEL,
            h, h, DMODEL, DMODEL, 0);
        ln_kernel<<<ROWS, 256, 0, stream>>>(h, lfg + l * DMODEL, lfb + l * DMODEL,
                                            tmp, DMODEL, 1e-5f);
        gemm_wmma_kernel<<<dim3(ROWS / 16, 8), 128, 0, stream>>>(
            tmp, ff1_w + (size_t)l * DMODEL * FFDIM, ff1_b + l * FFDIM,
            nullptr, big, DMODEL, FFDIM, 1 /* gelu */);
        gemm_wmma_kernel<<<dim3(ROWS / 16, 2), 128, 0, stream>>>(
            big, ff2_w + (size_t)l * FFDIM * DMODEL, ff2_b + l * DMODEL,
            h, h, FFDIM, DMODEL, 0);
    }

    // back end: final LN -> output GEMM (N=128) -> time-major transpose
    ln_kernel<<<ROWS, 256, 0, stream>>>(h, fln_g, fln_b, tmp, DMODEL, 1e-5f);
    gemm_wmma_kernel<<<dim3(ROWS / 16, 1), 128, 0, stream>>>(
        tmp, op_w, op_b, nullptr, qkv, DMODEL, 128, 0);
    output_transpose_kernel<<<(ROWS * 128 + 255) / 256, 256, 0, stream>>>(qkv, out);
}